// FlowLMMainONNX_36197984371415
// MI455X (gfx1250) — compile-verified
//
#include <hip/hip_runtime.h>
#include <math.h>

#define D_MODEL   1024
#define NUM_HEADS 16
#define HEAD_DIM  64
#define NUM_LAYERS 16
#define LDIM      512
#define D_FF      4096
#define PAST_LEN  4096
#define T_KV      (PAST_LEN + 1)
#define LN_EPS    1e-5f
#define NSPLIT    32
#define ACHUNK    ((T_KV + NSPLIT - 1) / NSPLIT)   /* 129 */

typedef __attribute__((ext_vector_type(2))) float v2f;
typedef __attribute__((ext_vector_type(8))) float v8f;

/* ---- workspace layout (float offsets) ---- */
#define OFF_X      0
#define OFF_XN     1024
#define OFF_QKV    2048
#define OFF_Q      5120
#define OFF_KNEW   6144
#define OFF_VNEW   7168
#define OFF_ATT    8192
#define OFF_H      9216
#define OFF_SEQ    13312
#define OFF_PART   13824      /* up to 32768 floats */
#define OFF_APART  46592      /* 16*32*66 = 33792 floats */

/* ------------------------------------------------------------------ */
/* NaN fixup of the input sequence                                     */
__global__ void seqfix_kernel(const float* __restrict__ seq,
                              const float* __restrict__ bos,
                              float* __restrict__ o) {
  int i = blockIdx.x * blockDim.x + threadIdx.x;
  if (i < LDIM) { float v = seq[i]; o[i] = (v != v) ? bos[i] : v; }
}

/* ------------------------------------------------------------------ */
/* Copy past KV cache (L,2,4096,1024) -> out region (L,2,4097,1024)    */
__global__ void copy_cache_kernel(const float* __restrict__ src,
                                  float* __restrict__ dst) {
  size_t ls = blockIdx.y;                       /* 0..31 = (layer,kv) */
  const float* s = src + ls * ((size_t)PAST_LEN * D_MODEL);
  float*       d = dst + ls * ((size_t)T_KV   * D_MODEL);
  size_t stride = (size_t)gridDim.x * blockDim.x;
  size_t total  = (size_t)PAST_LEN * D_MODEL;
  for (size_t i = (size_t)blockIdx.x * blockDim.x + threadIdx.x; i < total; i += stride)
    d[i] = s[i];
}

/* ------------------------------------------------------------------ */
/* GEMV via V_WMMA_F32_16X16X4_F32, compile-time K-chunk for full      */
/* unrolling (no scalar division, straight-line load+wmma stream).     */
/* y[n0+c] partial = sum_k x[k] * W[k][n0+c] over this block's k-slice.*/
/* A frag (16x4 f32, only row M=0 live): vgpr0 = K0(lo)/K2(hi),        */
/*                                       vgpr1 = K1(lo)/K3(hi).        */
/* B frag (4x16 f32): vgpr0 = row K0(lo)/K2(hi), vgpr1 = K1(lo)/K3(hi).*/
template <int KCHUNK, int N>
__global__ void gemv_wmma_partial(const float* __restrict__ x,
                                  const float* __restrict__ W,
                                  float* __restrict__ part) {
  __shared__ float xs[KCHUNK];
  int tile = blockIdx.x;
  int ks   = blockIdx.y;
  int SK   = gridDim.y;
  int lane = threadIdx.x;          /* block = 32 (one wave) */
  int half = lane >> 4;            /* 0: lanes 0-15, 1: lanes 16-31 */
  int c    = lane & 15;
  int n0   = tile * 16;
  int k0   = ks * KCHUNK;

#pragma unroll
  for (int i = 0; i < KCHUNK / 32; ++i) xs[i * 32 + lane] = x[k0 + i * 32 + lane];
  __syncthreads();

  const float* Wb = W + (size_t)k0 * N + n0 + c;
  v8f acc = {};
#pragma unroll
  for (int kk = 0; kk < KCHUNK; kk += 4) {
    float xa = xs[kk + 2 * half];
    float xb = xs[kk + 2 * half + 1];
    v2f a; a.x = (c == 0) ? xa : 0.0f; a.y = (c == 0) ? xb : 0.0f;
    const float* Wr = Wb + (size_t)(kk + 2 * half) * N;
    v2f b; b.x = Wr[0]; b.y = Wr[N];
    if ((kk & 15) == 0)
      __builtin_prefetch(Wr + 16 * (size_t)N, 0, 1);  /* global_prefetch_b8 */
    acc = __builtin_amdgcn_wmma_f32_16x16x4_f32(false, a, false, b,
                                                (short)0, acc, false, false);
  }
  /* D row M=0 lives in acc[0] on lanes 0-15, column = lane */
  if (half == 0) part[((size_t)tile * SK + ks) * 16 + c] = acc[0];
}

/* sum split-K partials + bias, optional exact GELU, optional residual */
__global__ void gemv_finish(const float* __restrict__ part,
                            const float* __restrict__ bias,
                            const float* __restrict__ resid,
                            float* __restrict__ y,
                            int N, int SK, int act) {
  int n = blockIdx.x * blockDim.x + threadIdx.x;
  if (n >= N) return;
  int tile = n >> 4, c = n & 15;
  float s = bias[n];
  for (int k = 0; k < SK; ++k) s += part[((size_t)tile * SK + k) * 16 + c];
  if (act) s = 0.5f * s * (1.0f + erff(s * 0.70710678118654752f));
  if (resid) s += resid[n];
  y[n] = s;
}

/* ------------------------------------------------------------------ */
/* LayerNorm over 1024 elems, one block of 1024 threads (32 waves)     */
__global__ void ln_kernel(const float* __restrict__ x, const float* __restrict__ w,
                          const float* __restrict__ b, float* __restrict__ o) {
  __shared__ float red[32];
  int tid = threadIdx.x;
  float v = x[tid];
  float s = v;
  for (int of = 16; of; of >>= 1) s += __shfl_xor(s, of, 32);
  if ((tid & 31) == 0) red[tid >> 5] = s;
  __syncthreads();
  if (tid < 32) {
    float r = red[tid];
    for (int of = 16; of; of >>= 1) r += __shfl_xor(r, of, 32);
    if (tid == 0) red[0] = r;
  }
  __syncthreads();
  float mu = red[0] * (1.0f / 1024.0f);
  __syncthreads();
  float dv = v - mu;
  float q = dv * dv;
  for (int of = 16; of; of >>= 1) q += __shfl_xor(q, of, 32);
  if ((tid & 31) == 0) red[tid >> 5] = q;
  __syncthreads();
  if (tid < 32) {
    float r = red[tid];
    for (int of = 16; of; of >>= 1) r += __shfl_xor(r, of, 32);
    if (tid == 0) red[0] = r;
  }
  __syncthreads();
  float var = red[0] * (1.0f / 1024.0f);
  o[tid] = dv * rsqrtf(var + LN_EPS) * w[tid] + b[tid];
}

/* ------------------------------------------------------------------ */
/* RoPE q/k, scatter new k,v into ws and into the out cache at t=4096  */
__global__ void rope_scatter(const float* __restrict__ qkv,
                             const int* __restrict__ clen,
                             float* __restrict__ qout,
                             float* __restrict__ knew,
                             float* __restrict__ vnew,
                             float* __restrict__ kvout, int layer) {
  int tid = threadIdx.x;                 /* block = 1024 */
  float pos = (float)clen[0];
  size_t baseK = ((size_t)(layer * 2 + 0) * T_KV + PAST_LEN) * D_MODEL;
  size_t baseV = ((size_t)(layer * 2 + 1) * T_KV + PAST_LEN) * D_MODEL;
  float vv = qkv[2 * D_MODEL + tid];
  vnew[tid] = vv;
  kvout[baseV + tid] = vv;
  if (tid < 512) {
    int i = tid & 31;
    int base = (tid >> 5) * HEAD_DIM + 2 * i;
    float freq = __expf((float)i * (-0.28782313662425572f)); /* -2*ln(1e4)/64 */
    float ang = freq * pos;
    float cs = cosf(ang), sn = sinf(ang);
    float qr = qkv[base], qi = qkv[base + 1];
    float kr = qkv[D_MODEL + base], ki = qkv[D_MODEL + base + 1];
    qout[base]     = qr * cs - qi * sn;
    qout[base + 1] = qr * sn + qi * cs;
    float k0 = kr * cs - ki * sn, k1 = kr * sn + ki * cs;
    knew[base] = k0; knew[base + 1] = k1;
    kvout[baseK + base] = k0; kvout[baseK + base + 1] = k1;
  }
}

/* ------------------------------------------------------------------ */
/* Flash-decode partials: per (head, split): max, sumexp, sum p*V      */
__global__ void attn_partial(const float* __restrict__ q,
                             const float* __restrict__ kpast,
                             const float* __restrict__ vpast,
                             const float* __restrict__ knew,
                             const float* __restrict__ vnew,
                             float* __restrict__ part) {
  int h = blockIdx.x, sp = blockIdx.y;
  int tid = threadIdx.x;               /* block = 128 (4 waves) */
  int wave = tid >> 5, lane = tid & 31, half = lane >> 4, c = lane & 15;
  int t0 = sp * ACHUNK;
  int t1 = t0 + ACHUNK; if (t1 > T_KV) t1 = T_KV;
  __shared__ __align__(16) float qs[64];
  __shared__ float sc[192];
  __shared__ float red[32];
  if (tid < 64) qs[tid] = q[h * HEAD_DIM + tid];
  __syncthreads();

  /* scores: half-wave per row, 16 lanes x float4 = one 256B K row */
  for (int base = t0; base < t1; base += 8) {
    int r = base + wave * 2 + half;
    float p = 0.0f;
    if (r < t1) {
      const float* krow = (r < PAST_LEN)
          ? (kpast + (size_t)r * D_MODEL + h * HEAD_DIM)
          : (knew + h * HEAD_DIM);
      float4 kv4 = *(const float4*)(krow + c * 4);
      float4 q4  = *(const float4*)(qs + c * 4);
      p = kv4.x * q4.x + kv4.y * q4.y + kv4.z * q4.z + kv4.w * q4.w;
    }
    p += __shfl_xor(p, 1, 16);
    p += __shfl_xor(p, 2, 16);
    p += __shfl_xor(p, 4, 16);
    p += __shfl_xor(p, 8, 16);
    if (c == 0 && r < t1) sc[r - t0] = p * 0.125f;   /* 1/sqrt(64) */
  }
  __syncthreads();

  int n = t1 - t0;
  float m = -3.0e38f;
  for (int i = tid; i < n; i += 128) m = fmaxf(m, sc[i]);
  for (int of = 16; of; of >>= 1) m = fmaxf(m, __shfl_xor(m, of, 32));
  if (lane == 0) red[wave] = m;
  __syncthreads();
  m = fmaxf(fmaxf(red[0], red[1]), fmaxf(red[2], red[3]));
  __syncthreads();

  float ssum = 0.0f;
  for (int i = tid; i < n; i += 128) { float e = __expf(sc[i] - m); sc[i] = e; ssum += e; }
  for (int of = 16; of; of >>= 1) ssum += __shfl_xor(ssum, of, 32);
  if (lane == 0) red[wave] = ssum;
  __syncthreads();
  float S = red[0] + red[1] + red[2] + red[3];

  /* weighted V: two rows in parallel, 64 threads each (coalesced rows) */
  int d = tid & 63, tp = tid >> 6;
  float acc = 0.0f;
  for (int r = t0 + tp; r < t1; r += 2) {
    const float* vrow = (r < PAST_LEN)
        ? (vpast + (size_t)r * D_MODEL + h * HEAD_DIM)
        : (vnew + h * HEAD_DIM);
    acc += sc[r - t0] * vrow[d];
  }
  __syncthreads();
  sc[tid] = acc;
  __syncthreads();
  float* pp = part + ((size_t)h * NSPLIT + sp) * 66;
  if (tid < 64) pp[2 + tid] = sc[tid] + sc[tid + 64];
  if (tid == 0) { pp[0] = m; pp[1] = S; }
}

__global__ void attn_combine(const float* __restrict__ part,
                             float* __restrict__ out) {
  int h = blockIdx.x, tid = threadIdx.x;   /* block = 64 */
  float M = -3.0e38f;
  for (int s = 0; s < NSPLIT; ++s)
    M = fmaxf(M, part[((size_t)h * NSPLIT + s) * 66]);
  float S = 0.0f, acc = 0.0f;
  for (int s = 0; s < NSPLIT; ++s) {
    const float* pp = part + ((size_t)h * NSPLIT + s) * 66;
    float w = __expf(pp[0] - M);
    S   += w * pp[1];
    acc += w * pp[2 + tid];
  }
  out[h * HEAD_DIM + tid] = acc / S;
}

/* ------------------------------------------------------------------ */
/* Final LN + eos logit + cache_len output                              */
__global__ void final_kernel(const float* __restrict__ x, const float* __restrict__ w,
                             const float* __restrict__ b, const float* __restrict__ weos,
                             const float* __restrict__ beos, const int* __restrict__ clen,
                             float* __restrict__ out) {
  __shared__ float red[32];
  int tid = threadIdx.x;                   /* block = 1024 */
  float v = x[tid];
  float s = v;
  for (int of = 16; of; of >>= 1) s += __shfl_xor(s, of, 32);
  if ((tid & 31) == 0) red[tid >> 5] = s;
  __syncthreads();
  if (tid < 32) {
    float r = red[tid];
    for (int of = 16; of; of >>= 1) r += __shfl_xor(r, of, 32);
    if (tid == 0) red[0] = r;
  }
  __syncthreads();
  float mu = red[0] * (1.0f / 1024.0f);
  __syncthreads();
  float dv = v - mu;
  float q = dv * dv;
  for (int of = 16; of; of >>= 1) q += __shfl_xor(q, of, 32);
  if ((tid & 31) == 0) red[tid >> 5] = q;
  __syncthreads();
  if (tid < 32) {
    float r = red[tid];
    for (int of = 16; of; of >>= 1) r += __shfl_xor(r, of, 32);
    if (tid == 0) red[0] = r;
  }
  __syncthreads();
  float var = red[0] * (1.0f / 1024.0f);
  float cond = dv * rsqrtf(var + LN_EPS) * w[tid] + b[tid];
  out[tid] = cond;
  __syncthreads();
  float e = cond * weos[tid];
  for (int of = 16; of; of >>= 1) e += __shfl_xor(e, of, 32);
  if ((tid & 31) == 0) red[tid >> 5] = e;
  __syncthreads();
  if (tid == 0) {
    float r = 0.0f;
    for (int i = 0; i < 32; ++i) r += red[i];
    out[D_MODEL] = r + beos[0];
    out[(size_t)1025 + (size_t)NUM_LAYERS * 2 * T_KV * D_MODEL] =
        (float)(clen[0] + 1);
  }
}

/* ================================================================== */
extern "C" void kernel_launch(void* const* d_in, const int* in_sizes, int n_in,
                              void* d_out, int out_size, void* d_ws, size_t ws_size,
                              hipStream_t stream) {
  const float* seq   = (const float*)d_in[0];
  /* d_in[1] = text_embeddings, empty */
  const float* kvc   = (const float*)d_in[2];
  const float* bos   = (const float*)d_in[3];
  const float* W_in  = (const float*)d_in[4];
  const float* b_in  = (const float*)d_in[5];
  const float* ln1w  = (const float*)d_in[6];
  const float* ln1b  = (const float*)d_in[7];
  const float* Wqkv  = (const float*)d_in[8];
  const float* bqkv  = (const float*)d_in[9];
  const float* Wo    = (const float*)d_in[10];
  const float* bo    = (const float*)d_in[11];
  const float* ln2w  = (const float*)d_in[12];
  const float* ln2b  = (const float*)d_in[13];
  const float* W1    = (const float*)d_in[14];
  const float* b1    = (const float*)d_in[15];
  const float* W2    = (const float*)d_in[16];
  const float* b2    = (const float*)d_in[17];
  const float* onw   = (const float*)d_in[18];
  const float* onb   = (const float*)d_in[19];
  const float* Weos  = (const float*)d_in[20];
  const float* beos  = (const float*)d_in[21];
  const int*   clen  = (const int*)d_in[22];

  float* out   = (float*)d_out;
  float* ws    = (float*)d_ws;
  float* kvout = out + 1025;   /* new_kv_cache region */

  /* past-cache copy into the output (independent of layer compute) */
  copy_cache_kernel<<<dim3(1024, 32), 256, 0, stream>>>(kvc, kvout);

  /* input projection */
  seqfix_kernel<<<2, 256, 0, stream>>>(seq, bos, ws + OFF_SEQ);
  gemv_wmma_partial<128, D_MODEL><<<dim3(64, 4), 32, 0, stream>>>(
      ws + OFF_SEQ, W_in, ws + OFF_PART);
  gemv_finish<<<4, 256, 0, stream>>>(ws + OFF_PART, b_in, nullptr,
                                     ws + OFF_X, D_MODEL, 4, 0);

  for (int l = 0; l < NUM_LAYERS; ++l) {
    ln_kernel<<<1, 1024, 0, stream>>>(ws + OFF_X, ln1w + l * D_MODEL,
                                      ln1b + l * D_MODEL, ws + OFF_XN);
    /* qkv projection: 1024 x 3072 */
    gemv_wmma_partial<128, 3 * D_MODEL><<<dim3(192, 8), 32, 0, stream>>>(
        ws + OFF_XN, Wqkv + (size_t)l * D_MODEL * 3 * D_MODEL, ws + OFF_PART);
    gemv_finish<<<12, 256, 0, stream>>>(ws + OFF_PART, bqkv + l * 3 * D_MODEL,
                                        nullptr, ws + OFF_QKV, 3 * D_MODEL, 8, 0);
    rope_scatter<<<1, 1024, 0, stream>>>(ws + OFF_QKV, clen, ws + OFF_Q,
                                         ws + OFF_KNEW, ws + OFF_VNEW, kvout, l);
    attn_partial<<<dim3(NUM_HEADS, NSPLIT), 128, 0, stream>>>(
        ws + OFF_Q,
        kvc + (size_t)(l * 2 + 0) * PAST_LEN * D_MODEL,
        kvc + (size_t)(l * 2 + 1) * PAST_LEN * D_MODEL,
        ws + OFF_KNEW, ws + OFF_VNEW, ws + OFF_APART);
    attn_combine<<<NUM_HEADS, 64, 0, stream>>>(ws + OFF_APART, ws + OFF_ATT);
    /* output projection + residual */
    gemv_wmma_partial<128, D_MODEL><<<dim3(64, 8), 32, 0, stream>>>(
        ws + OFF_ATT, Wo + (size_t)l * D_MODEL * D_MODEL, ws + OFF_PART);
    gemv_finish<<<4, 256, 0, stream>>>(ws + OFF_PART, bo + l * D_MODEL,
                                       ws + OFF_X, ws + OFF_X, D_MODEL, 8, 0);
    /* MLP */
    ln_kernel<<<1, 1024, 0, stream>>>(ws + OFF_X, ln2w + l * D_MODEL,
                                      ln2b + l * D_MODEL, ws + OFF_XN);
    gemv_wmma_partial<128, D_FF><<<dim3(256, 8), 32, 0, stream>>>(
        ws + OFF_XN, W1 + (size_t)l * D_MODEL * D_FF, ws + OFF_PART);
    gemv_finish<<<16, 256, 0, stream>>>(ws + OFF_PART, b1 + l * D_FF,
                                        nullptr, ws + OFF_H, D_FF, 8, 1);
    gemv_wmma_partial<256, D_MODEL><<<dim3(64, 16), 32, 0, stream>>>(
        ws + OFF_H, W2 + (size_t)l * D_FF * D_MODEL, ws + OFF_PART);
    gemv_finish<<<4, 256, 0, stream>>>(ws + OFF_PART, b2 + l * D_MODEL,
                                       ws + OFF_X, ws + OFF_X, D_MODEL, 16, 0);
  }

  final_kernel<<<1, 1024, 0, stream>>>(ws + OFF_X, onw, onb, Weos, beos, clen, out);
}